// SimpleYOLOLoss_15736760172790
// MI455X (gfx1250) — compile-verified
//
#include <hip/hip_runtime.h>
#include <stdint.h>

typedef __attribute__((ext_vector_type(2))) float v2f;
typedef __attribute__((ext_vector_type(8))) float v8f;

// ---------------------------------------------------------------------------
// Wave(32) sum using V_WMMA_F32_16X16X4_F32 as an exact f32 reduction:
// a = {x, 0}, B = all ones -> D[m][n] = x_m + x_{m+16} (+C=0), exact f32.
// Lanes 0-15 hold rows 0-7 in their 8 D vgprs, lanes 16-31 rows 8-15;
// sum of the 8 components + shfl_xor(16) gives the full 32-lane sum.
// ---------------------------------------------------------------------------
__device__ __forceinline__ float wave_sum(float x) {
#if __has_builtin(__builtin_amdgcn_wmma_f32_16x16x4_f32)
  v2f a; a.x = x;    a.y = 0.0f;
  v2f b; b.x = 1.0f; b.y = 1.0f;
  v8f c = {0.0f, 0.0f, 0.0f, 0.0f, 0.0f, 0.0f, 0.0f, 0.0f};
  c = __builtin_amdgcn_wmma_f32_16x16x4_f32(false, a, false, b, (short)0, c,
                                            false, false);
  float s = c[0] + c[1] + c[2] + c[3] + c[4] + c[5] + c[6] + c[7];
  s += __shfl_xor(s, 16, 32);
  return s;
#else
  for (int off = 16; off > 0; off >>= 1) x += __shfl_xor(x, off, 32);
  return x;
#endif
}

// Block (256 threads = 8 waves) sum; result valid on wave 0 (incl. thread 0).
__device__ __forceinline__ float block_sum(float v, float* red) {
  float w = wave_sum(v);
  const int lane = threadIdx.x & 31;
  const int wid  = threadIdx.x >> 5;
  if (lane == 0) red[wid] = w;
  __syncthreads();
  float r = (threadIdx.x < 8) ? red[threadIdx.x] : 0.0f;
  r = wave_sum(r);
  __syncthreads();  // safe reuse of `red`
  return r;
}

// ---------------------------------------------------------------------------
// CDNA5 async global->LDS copy (ASYNCcnt-tracked), GV mode (64-bit vaddr).
// ---------------------------------------------------------------------------
__device__ __forceinline__ void async_load_b128(uint32_t lds_byte_addr,
                                                const void* gptr) {
  asm volatile("global_load_async_to_lds_b128 %0, %1, off"
               :: "v"(lds_byte_addr), "v"((uint64_t)(uintptr_t)gptr)
               : "memory");
}
__device__ __forceinline__ void wait_async0() {
  asm volatile("s_wait_asynccnt 0x0" ::: "memory");
}

// Problem constants (fixed by setup_inputs)
#define BATCH 32
#define NTGT  64
#define NC    4
#define RM    16
// rows per level = B*HW ; cumulative boundaries (multiples of 256)
#define ROWS0 524288   // 32*128*128
#define CUM1  655360   // +32*64*64
#define CUM2  688128   // +32*32*32
#define CUM3  696320   // +32*16*16
#define CLS_BLOCKS 2720  // CUM3 / 256

// Exact reciprocals (all divisors are powers of two -> bit-exact vs division)
__device__ __forceinline__ float inv_hw(int level) {
  return (level == 0) ? (1.0f / 16384.0f)
       : (level == 1) ? (1.0f / 4096.0f)
       : (level == 2) ? (1.0f / 1024.0f)
       :                (1.0f / 256.0f);
}
__device__ __forceinline__ float inv_stride(int level) {
  return (level == 0) ? (1.0f / 8.0f)
       : (level == 1) ? (1.0f / 16.0f)
       : (level == 2) ? (1.0f / 32.0f)
       :                (1.0f / 64.0f);
}

// ---------------------------------------------------------------------------
// Kernel 0: nvalid[b] = #targets with class_id >= 0
// ---------------------------------------------------------------------------
__global__ void nvalid_kernel(const float* __restrict__ tg,
                              float* __restrict__ nv) {
  int b = threadIdx.x;
  if (b < BATCH) {
    float cnt = 0.0f;
    for (int n = 0; n < NTGT; ++n)
      cnt += (tg[(size_t)(b * NTGT + n) * 5] >= 0.0f) ? 1.0f : 0.0f;
    nv[b] = cnt;
  }
}

// ---------------------------------------------------------------------------
// Kernel 1: streaming cls logsumexp. One logits "row" (4 contiguous floats of
// the [C,H,W] buffer, per the faithful reshape) per lane, brought in via the
// async-to-LDS path. Accumulates (lse - x0) * nvalid[b] / HW per block.
// ---------------------------------------------------------------------------
__global__ void __launch_bounds__(256)
cls_base_kernel(const float* __restrict__ c0, const float* __restrict__ c1,
                const float* __restrict__ c2, const float* __restrict__ c3,
                const float* __restrict__ nv, float* __restrict__ clsPart) {
  __shared__ float smem[256 * 4];
  __shared__ float red[8];

  int gid = blockIdx.x * 256 + threadIdx.x;
  int level, local;
  if (gid < ROWS0)      { level = 0; local = gid; }
  else if (gid < CUM1)  { level = 1; local = gid - ROWS0; }
  else if (gid < CUM2)  { level = 2; local = gid - CUM1; }
  else                  { level = 3; local = gid - CUM2; }

  const int W  = 128 >> level;
  const int HW = W * W;
  const float invHW = inv_hw(level);
  const float* cp = (level == 0) ? c0 : (level == 1) ? c1
                   : (level == 2) ? c2 : c3;
  const int b = local / HW;          // HW is a power of two -> cheap
  const int r = local - b * HW;
  const float* src = cp + ((size_t)b * HW + (size_t)r) * 4;  // = b*4*HW + r*4

  uint32_t lds = (uint32_t)(uintptr_t)(&smem[threadIdx.x * 4]);
  async_load_b128(lds, src);
  wait_async0();

  const float x0 = smem[threadIdx.x * 4 + 0];
  const float x1 = smem[threadIdx.x * 4 + 1];
  const float x2 = smem[threadIdx.x * 4 + 2];
  const float x3 = smem[threadIdx.x * 4 + 3];
  const float m  = fmaxf(fmaxf(x0, x1), fmaxf(x2, x3));
  const float lse = m + __logf(__expf(x0 - m) + __expf(x1 - m) +
                               __expf(x2 - m) + __expf(x3 - m));
  const float val = (lse - x0) * (nv[b] * invHW);

  const float t = block_sum(val, red);
  if (threadIdx.x == 0) clsPart[blockIdx.x] = t;
}

// ---------------------------------------------------------------------------
// Kernel 2: one thread per (level, b, n) target item: adj + box + dfl terms.
// 8192 threads = 32 blocks x 256.
// ---------------------------------------------------------------------------
__global__ void __launch_bounds__(256)
target_kernel(const float* __restrict__ c0, const float* __restrict__ c1,
              const float* __restrict__ c2, const float* __restrict__ c3,
              const float* __restrict__ r0, const float* __restrict__ r1,
              const float* __restrict__ r2, const float* __restrict__ r3,
              const float* __restrict__ tg,
              float* __restrict__ adjP, float* __restrict__ boxP,
              float* __restrict__ dflP) {
  __shared__ float red[8];
  const int gid   = blockIdx.x * 256 + threadIdx.x;  // 0..8191
  const int level = gid >> 11;                       // / (32*64)
  const int rem   = gid & 2047;
  const int b     = rem >> 6;
  const int n     = rem & 63;

  const int   W  = 128 >> level;
  const int   HW = W * W;
  const float sF   = (float)(8 << level);
  const float invS = inv_stride(level);   // exact: 1/8,1/16,1/32,1/64
  const float invHW = inv_hw(level);
  const float* cp = (level == 0) ? c0 : (level == 1) ? c1
                   : (level == 2) ? c2 : c3;
  const float* rp = (level == 0) ? r0 : (level == 1) ? r1
                   : (level == 2) ? r2 : r3;

  const float* tgp = tg + (size_t)(b * NTGT + n) * 5;
  const float cidf = tgp[0];
  const bool  valid = (cidf >= 0.0f);
  const int   cid = (int)fmaxf(cidf, 0.0f);
  const float x1 = tgp[1], y1 = tgp[2], x2 = tgp[3], y2 = tgp[4];
  const float cx = (x1 + x2) * 0.5f;
  const float cy = (y1 + y2) * 0.5f;

  int gx = (int)(cx * invS);  gx = gx < 0 ? 0 : (gx > W - 1 ? W - 1 : gx);
  int gy = (int)(cy * invS);  gy = gy < 0 ? 0 : (gy > W - 1 ? W - 1 : gy);
  const int t = gy * W + gx;

  // adj uses the faithful reshape indexing: flat offset t*NC + c in [C,H,W]
  const float* cb = cp + (size_t)b * NC * HW;
  float adj = (cb[(size_t)t * 4] - cb[(size_t)t * 4 + cid]) * invHW;

  const float* rb = rp + (size_t)b * (4 * RM) * HW + t;
  const float tre[4] = { (cx - (float)gx * sF) * invS,
                         (cy - (float)gy * sF) * invS,
                         (x2 - x1) * invS,
                         (y2 - y1) * invS };
  float box = 0.0f, dfl = 0.0f;
#pragma unroll
  for (int sd = 0; sd < 4; ++sd) {
    float mean = 0.0f, ad = 0.0f;
    const float trv = tre[sd];
#pragma unroll
    for (int j = 0; j < RM; ++j) {
      const float p = rb[(size_t)(sd * RM + j) * HW];
      mean += p;
      ad   += fabsf(p - trv);
    }
    mean *= (1.0f / (float)RM);
    box += fabsf(mean - trv);
    dfl += ad;
  }
  box *= 0.25f;
  dfl *= (1.0f / 64.0f);
  if (!valid) { adj = 0.0f; box = 0.0f; dfl = 0.0f; }

  float s1 = block_sum(adj, red);
  if (threadIdx.x == 0) adjP[blockIdx.x] = s1;
  float s2 = block_sum(box, red);
  if (threadIdx.x == 0) boxP[blockIdx.x] = s2;
  float s3 = block_sum(dfl, red);
  if (threadIdx.x == 0) dflP[blockIdx.x] = s3;
}

// ---------------------------------------------------------------------------
// Kernel 3: reduce partials and emit (total, cls, box, dfl).
// ---------------------------------------------------------------------------
__global__ void __launch_bounds__(256)
finalize_kernel(const float* __restrict__ clsPart,
                const float* __restrict__ adjP,
                const float* __restrict__ boxP,
                const float* __restrict__ dflP,
                float* __restrict__ out) {
  __shared__ float red[8];
  float s = 0.0f;
  for (int i = threadIdx.x; i < CLS_BLOCKS; i += 256) s += clsPart[i];
  const float clsBase = block_sum(s, red);
  const float adj  = block_sum(threadIdx.x < 32 ? adjP[threadIdx.x] : 0.0f, red);
  const float boxS = block_sum(threadIdx.x < 32 ? boxP[threadIdx.x] : 0.0f, red);
  const float dflS = block_sum(threadIdx.x < 32 ? dflP[threadIdx.x] : 0.0f, red);
  if (threadIdx.x == 0) {
    const float cls = clsBase + adj;
    out[0] = 0.3f * cls + 8.0f * boxS + 1.5f * dflS;
    out[1] = cls;
    out[2] = boxS;
    out[3] = dflS;
  }
}

// ---------------------------------------------------------------------------
extern "C" void kernel_launch(void* const* d_in, const int* in_sizes, int n_in,
                              void* d_out, int out_size, void* d_ws,
                              size_t ws_size, hipStream_t stream) {
  (void)in_sizes; (void)n_in; (void)out_size; (void)ws_size;
  const float* c0 = (const float*)d_in[0];
  const float* c1 = (const float*)d_in[1];
  const float* c2 = (const float*)d_in[2];
  const float* c3 = (const float*)d_in[3];
  const float* r0 = (const float*)d_in[4];
  const float* r1 = (const float*)d_in[5];
  const float* r2 = (const float*)d_in[6];
  const float* r3 = (const float*)d_in[7];
  const float* tg = (const float*)d_in[8];

  float* ws      = (float*)d_ws;
  float* nv      = ws;                    // 32
  float* clsPart = ws + 32;               // 2720
  float* adjP    = clsPart + CLS_BLOCKS;  // 32
  float* boxP    = adjP + 32;             // 32
  float* dflP    = boxP + 32;             // 32

  nvalid_kernel<<<1, 64, 0, stream>>>(tg, nv);
  cls_base_kernel<<<CLS_BLOCKS, 256, 0, stream>>>(c0, c1, c2, c3, nv, clsPart);
  target_kernel<<<32, 256, 0, stream>>>(c0, c1, c2, c3, r0, r1, r2, r3, tg,
                                        adjP, boxP, dflP);
  finalize_kernel<<<1, 256, 0, stream>>>(clsPart, adjP, boxP, dflP,
                                         (float*)d_out);
}